// RoBERTaForTopKEntailment_45389214384546
// MI455X (gfx1250) — compile-verified
//
#include <hip/hip_runtime.h>
#include <hip/hip_bf16.h>

#define B_ 64
#define S_ 512
#define H_ 1024
#define K_ 8
#define BK_ (B_ * K_)   // 512
#define LN_EPS 1e-5f

typedef float v2f __attribute__((ext_vector_type(2)));
typedef float v8f __attribute__((ext_vector_type(8)));

// ---------------------------------------------------------------------------
// Kernel 0: per (b,k) find first/last occurrence of segment k, derive the
// inner slice [start+1, end) and its count (matching hidden[start+1:end]).
// ---------------------------------------------------------------------------
__global__ void seg_ranges_kernel(const int* __restrict__ seg,
                                  int* __restrict__ lo,
                                  int* __restrict__ hi,
                                  int* __restrict__ cnt) {
    int i = threadIdx.x;              // 0..511, one per (b,k)
    if (i >= BK_) return;
    int b = i >> 3;                   // /K_
    int k = i & 7;                    // %K_
    const int* s = seg + b * S_;
    int mn = S_, mx = -1;
    for (int p = 0; p < S_; ++p) {
        int v = s[p];
        if (v == k) { if (p < mn) mn = p; mx = p; }
    }
    int l = mn + 1;                   // start+1
    int e = mx;                       // end (exclusive upper bound of inner)
    int c = e - l;                    // number of inner positions
    lo[i]  = l;
    hi[i]  = e;
    cnt[i] = (c > 0) ? c : 1;         // guard divide; degenerate segs -> 0
}

// ---------------------------------------------------------------------------
// Kernel 1: segment mean pooling (HBM roofline phase: ~124 MB streamed).
// One block per (b,k); 256 threads x float4 = full 4 KB row per iteration.
// b128 loads, coalesced; prefetch 4 rows ahead.
// ---------------------------------------------------------------------------
__global__ void seg_pool_kernel(const float* __restrict__ hs,
                                const int* __restrict__ lo,
                                const int* __restrict__ hi,
                                const int* __restrict__ cnt,
                                float* __restrict__ pooled) {
    int bk  = blockIdx.x;                          // 0..511
    int col = threadIdx.x * 4;                     // 0..1020
    int b   = bk >> 3;
    int l   = lo[bk];
    int e   = hi[bk];
    const float* base = hs + (size_t)b * S_ * H_ + col;
    float s0 = 0.f, s1 = 0.f, s2 = 0.f, s3 = 0.f;
    for (int p = l; p < e; ++p) {
        __builtin_prefetch(base + (size_t)(p + 4) * H_, 0, 0); // global_prefetch_b8
        const float4 v = *(const float4*)(base + (size_t)p * H_); // global_load_b128
        s0 += v.x; s1 += v.y; s2 += v.z; s3 += v.w;
    }
    float inv = 1.0f / (float)cnt[bk];
    float4 r = make_float4(s0 * inv, s1 * inv, s2 * inv, s3 * inv);
    *(float4*)(pooled + (size_t)bk * H_ + col) = r;
}

// ---------------------------------------------------------------------------
// Kernel 2: h = relu(pooled @ W1 + b1) via V_WMMA_F32_16X16X4_F32.
// 32x32 output tile per wave32 (2x2 WMMA fragments): per 4-wide k-step,
// 2 A-fragment loads + 2 B-fragment loads feed 4 WMMA ops -> halves L2
// traffic vs. 16x16 tiles (A: 64 MB, B: 64 MB total, all L2-resident).
// Layouts per ISA:
//   A (16x4 f32): lanes 0-15 row M=lane, K=k0,k0+1; lanes 16-31 K=k0+2,k0+3.
//   B (4x16 f32): mirrored (lane = column N).
//   C/D (16x16 f32, 8 VGPRs): VGPR i -> M=i (+8 for lanes 16-31), N=lane&15.
// ---------------------------------------------------------------------------
__global__ void gemm_relu_wmma_kernel(const float* __restrict__ pooled,
                                      const float* __restrict__ W1,
                                      const float* __restrict__ b1,
                                      float* __restrict__ h) {
    const int lane    = threadIdx.x & 31;
    const int wave    = threadIdx.x >> 5;
    const int tile    = blockIdx.x * (blockDim.x >> 5) + wave;
    const int ntilesN = H_ / 32;                   // 32
    const int tm      = tile / ntilesN;            // 0..15
    const int tn      = tile % ntilesN;            // 0..31
    const int row0    = tm * 32;
    const int col0    = tn * 32;
    const int half    = lane >> 4;                 // 0 or 1
    const int l       = lane & 15;

    const float* arow0 = pooled + (size_t)(row0 + l) * H_;        // A frag 0: M=l
    const float* arow1 = pooled + (size_t)(row0 + 16 + l) * H_;   // A frag 1: M=16+l
    const float* bcol0 = W1 + (col0 + l);                         // B frag 0: N=l
    const float* bcol1 = W1 + (col0 + 16 + l);                    // B frag 1: N=16+l

    v8f acc00 = {}, acc01 = {}, acc10 = {}, acc11 = {};
#pragma unroll 2
    for (int k0 = 0; k0 < H_; k0 += 4) {
        const int ka = k0 + half * 2;
        v2f a0, a1, b0, b1v;
        a0.x  = arow0[ka];            a0.y  = arow0[ka + 1];
        a1.x  = arow1[ka];            a1.y  = arow1[ka + 1];
        b0.x  = bcol0[(size_t)ka * H_];
        b0.y  = bcol0[(size_t)(ka + 1) * H_];
        b1v.x = bcol1[(size_t)ka * H_];
        b1v.y = bcol1[(size_t)(ka + 1) * H_];
        acc00 = __builtin_amdgcn_wmma_f32_16x16x4_f32(false, a0, false, b0,  (short)0, acc00, false, false);
        acc01 = __builtin_amdgcn_wmma_f32_16x16x4_f32(false, a0, false, b1v, (short)0, acc01, false, false);
        acc10 = __builtin_amdgcn_wmma_f32_16x16x4_f32(false, a1, false, b0,  (short)0, acc10, false, false);
        acc11 = __builtin_amdgcn_wmma_f32_16x16x4_f32(false, a1, false, b1v, (short)0, acc11, false, false);
    }

    const float bias0 = b1[col0 + l];
    const float bias1 = b1[col0 + 16 + l];
#pragma unroll
    for (int i = 0; i < 8; ++i) {
        const int m0 = row0 + i + half * 8;       // rows of acc0x
        const int m1 = m0 + 16;                   // rows of acc1x
        h[(size_t)m0 * H_ + col0 + l]      = fmaxf(acc00[i] + bias0, 0.0f);
        h[(size_t)m0 * H_ + col0 + 16 + l] = fmaxf(acc01[i] + bias1, 0.0f);
        h[(size_t)m1 * H_ + col0 + l]      = fmaxf(acc10[i] + bias0, 0.0f);
        h[(size_t)m1 * H_ + col0 + 16 + l] = fmaxf(acc11[i] + bias1, 0.0f);
    }
}

// ---------------------------------------------------------------------------
// Kernel 3: per-row LayerNorm + dot with W2 + b2. One 256-thread block per
// row (8 wave32s); shuffle + LDS block reductions.
// ---------------------------------------------------------------------------
__device__ __forceinline__ float wave_sum(float v) {
    for (int o = 16; o > 0; o >>= 1) v += __shfl_down(v, o, 32);
    return v;
}

__device__ __forceinline__ float block_sum(float v, float* sh) {
    int lane = threadIdx.x & 31;
    int w    = threadIdx.x >> 5;
    v = wave_sum(v);
    __syncthreads();                 // protect sh reuse across calls
    if (lane == 0) sh[w] = v;
    __syncthreads();
    float r = (threadIdx.x < 8) ? sh[threadIdx.x] : 0.0f;
    if (w == 0) {
        r = wave_sum(r);
        if (lane == 0) sh[0] = r;
    }
    __syncthreads();
    return sh[0];
}

__global__ void ln_head_kernel(const float* __restrict__ h,
                               const float* __restrict__ gamma,
                               const float* __restrict__ beta,
                               const float* __restrict__ W2,
                               const float* __restrict__ b2,
                               float* __restrict__ out) {
    __shared__ float sh[8];
    int row = blockIdx.x;                          // 0..511
    const float* hr = h + (size_t)row * H_;

    float x[4];
    float s = 0.0f, ss = 0.0f;
#pragma unroll
    for (int j = 0; j < 4; ++j) {
        int i = threadIdx.x + j * 256;
        x[j] = hr[i];
        s  += x[j];
        ss += x[j] * x[j];
    }
    float tot  = block_sum(s, sh);
    float tot2 = block_sum(ss, sh);
    float mean = tot * (1.0f / H_);
    float var  = tot2 * (1.0f / H_) - mean * mean;
    float rs   = rsqrtf(var + LN_EPS);

    float dot = 0.0f;
#pragma unroll
    for (int j = 0; j < 4; ++j) {
        int i = threadIdx.x + j * 256;
        float n = (x[j] - mean) * rs * gamma[i] + beta[i];
        dot += n * W2[i];
    }
    float d = block_sum(dot, sh);
    if (threadIdx.x == 0) out[row] = d + b2[0];
}

// ---------------------------------------------------------------------------
extern "C" void kernel_launch(void* const* d_in, const int* in_sizes, int n_in,
                              void* d_out, int out_size, void* d_ws, size_t ws_size,
                              hipStream_t stream) {
    const float* hs    = (const float*)d_in[0];
    const int*   seg   = (const int*)d_in[1];
    const float* W1    = (const float*)d_in[2];
    const float* b1    = (const float*)d_in[3];
    const float* gamma = (const float*)d_in[4];
    const float* beta  = (const float*)d_in[5];
    const float* W2    = (const float*)d_in[6];
    const float* b2    = (const float*)d_in[7];
    float*       out   = (float*)d_out;

    char* ws = (char*)d_ws;
    int*   lo     = (int*)ws;            // 512 ints
    int*   hi     = lo + BK_;            // 512 ints
    int*   cnt    = hi + BK_;            // 512 ints
    float* pooled = (float*)(ws + 8192);             // 512*1024 f32 = 2 MB
    float* hbuf   = pooled + (size_t)BK_ * H_;       // 512*1024 f32 = 2 MB

    seg_ranges_kernel<<<1, 512, 0, stream>>>(seg, lo, hi, cnt);
    seg_pool_kernel<<<BK_, 256, 0, stream>>>(hs, lo, hi, cnt, pooled);
    // 32x32 wave tiles: (512/32)*(1024/32) = 512 waves; 8 waves per block
    gemm_relu_wmma_kernel<<<512 / 8, 256, 0, stream>>>(pooled, W1, b1, hbuf);
    ln_head_kernel<<<BK_, 256, 0, stream>>>(hbuf, gamma, beta, W2, b2, out);
}